// SEGNN_55525337203318
// MI455X (gfx1250) — compile-verified
//
#include <hip/hip_runtime.h>
#include <math.h>

#define NNODES 4096
#define NEDGES 32768
#define NORM_E 32767.0f
#define INV_SQRT3 0.57735026918962576f
#define SQRT3F 1.7320508075688772f

typedef float v2f __attribute__((ext_vector_type(2)));
typedef float v8f __attribute__((ext_vector_type(8)));

// ---- CDNA5 WMMA: D(16x16,f32) = A(16x4,f32) x B(4x16,f32) + C ----
// A: lanes 0-15 hold M=lane, VGPR j -> K = 2*(lane>>4)+j.
// B: VGPR j -> K = 2*(lane>>4)+j, N = lane&15.
// C/D: VGPR v -> M = v + 8*(lane>>4), N = lane&15.
__device__ __forceinline__ v8f wmma4(float a0, float a1, float b0, float b1, v8f c) {
  v2f a; a[0] = a0; a[1] = a1;
  v2f b; b[0] = b0; b[1] = b1;
  return __builtin_amdgcn_wmma_f32_16x16x4_f32(false, a, false, b, (short)0, c,
                                               false, false);
}

__device__ __forceinline__ float gelu_f(float x) {
  const float c = 0.7978845608028654f;  // sqrt(2/pi), jax approximate=True
  float t = tanhf(c * (x + 0.044715f * x * x * x));
  return 0.5f * x * (1.0f + t);
}
__device__ __forceinline__ float sigm_f(float x) { return 1.0f / (1.0f + expf(-x)); }

// =====================  utility kernels  =====================
__global__ void init_copy(const float* __restrict__ ns, const float* __restrict__ nv,
                          float* __restrict__ cs, float* __restrict__ cv) {
  int i = blockIdx.x * blockDim.x + threadIdx.x;
  if (i < NNODES * 64) cs[i] = ns[i];
  if (i < NNODES * 48) cv[i] = nv[i];
}

__global__ void zero_buf(float* __restrict__ p, int n) {
  int i = blockIdx.x * blockDim.x + threadIdx.x;
  if (i < n) p[i] = 0.0f;
}

__global__ void writeout(const float* __restrict__ cs, const float* __restrict__ cv,
                         float* __restrict__ out) {
  int i = blockIdx.x * blockDim.x + threadIdx.x;
  if (i >= NNODES * 112) return;
  int n = i / 112, c = i % 112;
  out[i] = (c < 48) ? cv[n * 48 + c] : cs[n * 64 + (c - 48)];
}

__global__ void finalize_y(const float* __restrict__ sum_s, const float* __restrict__ sum_v,
                           const float* __restrict__ sum_av, const float* __restrict__ cnt,
                           float* __restrict__ ys, float* __restrict__ yv) {
  int n = blockIdx.x * blockDim.x + threadIdx.x;
  if (n >= NNODES) return;
  float c = cnt[n];
  float inv = 1.0f / (fmaxf(c, 1.0f) * NORM_E);
  for (int i = 0; i < 64; ++i) ys[n * 65 + i] = sum_s[n * 64 + i] * inv;
  ys[n * 65 + 64] = (c > 0.0f ? 1.0f : 0.0f) * (1.0f / NORM_E);
  for (int i = 0; i < 48; ++i) yv[n * 51 + i] = sum_v[n * 48 + i] * inv;
  for (int d = 0; d < 3; ++d) yv[n * 51 + 48 + d] = sum_av[n * 3 + d] * inv;
}

// =====================  edge kernel  =====================
// 16 edges per block, 256 threads = 8 waves.
// Waves 0-4: scalar out col tiles (80 cols). Waves 5-7: vector component d.
// K loops are split at the TP region boundary -> branchless uniform decode.
template <int M0, int M1>
__device__ __forceinline__ void edge_block_compute(
    int wid, int lane, const float (*SX)[129], const float (*VX)[97],
    const float (*AV)[3], float (*OS)[81], float (*OV)[49],
    const float* __restrict__ Ws, const float* __restrict__ Wv,
    const float* __restrict__ Bb) {
  const int mrow = lane & 15, hi = lane >> 4;
  if (wid < 5) {
    const int cb = wid * 16;
    const float* bp = Ws + cb + mrow;
    v8f acc = {0, 0, 0, 0, 0, 0, 0, 0};
#pragma unroll 4
    for (int k0 = 0; k0 < M0; k0 += 4) {            // ss region: a = sx[k]
      int k = k0 + 2 * hi;
      acc = wmma4(SX[mrow][k], SX[mrow][k + 1], bp[k * 80], bp[(k + 1) * 80], acc);
    }
#pragma unroll 4
    for (int k0 = M0; k0 < M0 + M1; k0 += 4) {      // dd region: a = (vx . a_v)/sqrt3
      int k = k0 + 2 * hi;
      int mm = (k - M0) * 3;
      float a0 = (VX[mrow][mm + 0] * AV[mrow][0] + VX[mrow][mm + 1] * AV[mrow][1] +
                  VX[mrow][mm + 2] * AV[mrow][2]) * INV_SQRT3;
      float a1 = (VX[mrow][mm + 3] * AV[mrow][0] + VX[mrow][mm + 4] * AV[mrow][1] +
                  VX[mrow][mm + 5] * AV[mrow][2]) * INV_SQRT3;
      acc = wmma4(a0, a1, bp[k * 80], bp[(k + 1) * 80], acc);
    }
#pragma unroll
    for (int v = 0; v < 8; ++v) OS[v + 8 * hi][cb + mrow] = acc[v] + Bb[cb + mrow];
  } else {
    const int d = wid - 5;
    const float* bp = Wv + mrow;
    v8f acc = {0, 0, 0, 0, 0, 0, 0, 0};
#pragma unroll 4
    for (int k0 = 0; k0 < M0; k0 += 4) {            // sv region: a = sx[k]*a_v[d]
      int k = k0 + 2 * hi;
      float av = AV[mrow][d];
      acc = wmma4(SX[mrow][k] * av, SX[mrow][k + 1] * av, bp[k * 16],
                  bp[(k + 1) * 16], acc);
    }
#pragma unroll 4
    for (int k0 = M0; k0 < M0 + M1; k0 += 4) {      // vs region: a = vx[k-M0][d]
      int k = k0 + 2 * hi;
      int mm = (k - M0) * 3 + d;
      acc = wmma4(VX[mrow][mm], VX[mrow][mm + 3], bp[k * 16], bp[(k + 1) * 16], acc);
    }
#pragma unroll
    for (int v = 0; v < 8; ++v) OV[v + 8 * hi][d * 16 + mrow] = acc[v];
  }
}

__global__ __launch_bounds__(256) void edge_kernel(
    const float* __restrict__ ns, const float* __restrict__ nv,
    const int* __restrict__ senders, const int* __restrict__ receivers,
    const float* __restrict__ e1_ws, const float* __restrict__ e1_b,
    const float* __restrict__ e1_wv, const float* __restrict__ e2_ws,
    const float* __restrict__ e2_b, const float* __restrict__ e2_wv,
    float* __restrict__ sum_s, float* __restrict__ sum_v,
    float* __restrict__ sum_av, float* __restrict__ cnt) {
  __shared__ float SX[16][129];   // scalar features, 128 used (odd stride: no bank conflict)
  __shared__ float VX[16][97];    // vector features flat m*3+d, 96 used
  __shared__ float AV[16][3];     // a_v = sqrt(3)*unit(r)
  __shared__ float OS[16][81];    // scalar GEMM out (pre-gate, + bias)
  __shared__ float OV[16][49];    // vector GEMM out flat d*16+m
  __shared__ int SIX[16], RIX[16];

  const int tid = threadIdx.x;
  const int lane = tid & 31, wid = tid >> 5;
  const int e0 = blockIdx.x * 16;

  if (tid < 16) { SIX[tid] = senders[e0 + tid]; RIX[tid] = receivers[e0 + tid]; }
  __syncthreads();
  for (int idx = tid; idx < 16 * 128; idx += 256) {   // m_s = concat(s_i, s_j)
    int r = idx >> 7, c = idx & 127;
    int node = (c < 64) ? SIX[r] : RIX[r];
    SX[r][c] = ns[node * 64 + (c & 63)];
  }
  for (int idx = tid; idx < 16 * 96; idx += 256) {    // m_v = concat(v_i, v_j)
    int r = idx / 96, c = idx % 96;
    int node = (c < 48) ? SIX[r] : RIX[r];
    VX[r][c] = nv[node * 48 + (c < 48 ? c : c - 48)];
  }
  __syncthreads();
  if (tid < 16) {
    float rx = VX[tid][0] - VX[tid][48];
    float ry = VX[tid][1] - VX[tid][49];
    float rz = VX[tid][2] - VX[tid][50];
    float rn = sqrtf(rx * rx + ry * ry + rz * rz);
    float inv = SQRT3F / fmaxf(rn, 1e-9f);
    AV[tid][0] = rx * inv; AV[tid][1] = ry * inv; AV[tid][2] = rz * inv;
  }
  __syncthreads();

#pragma unroll 1
  for (int blk = 0; blk < 3; ++blk) {
    const float* Ws = (blk == 0) ? e1_ws : e2_ws + (blk - 1) * 80 * 80;
    const float* Wv = (blk == 0) ? e1_wv : e2_wv + (blk - 1) * 80 * 16;
    const float* Bb = (blk == 0) ? e1_b : e2_b + (blk - 1) * 80;
    if (blk == 0)
      edge_block_compute<128, 32>(wid, lane, SX, VX, AV, OS, OV, Ws, Wv, Bb);
    else
      edge_block_compute<64, 16>(wid, lane, SX, VX, AV, OS, OV, Ws, Wv, Bb);
    __syncthreads();
    // gate: scal = gelu(s[:64]); v *= sigmoid(s[64:80])
    for (int idx = tid; idx < 16 * 64; idx += 256) {
      int r = idx >> 6, c = idx & 63;
      SX[r][c] = gelu_f(OS[r][c]);
    }
    for (int idx = tid; idx < 16 * 48; idx += 256) {
      int r = idx / 48, c = idx % 48;
      int m = c / 3, d = c % 3;
      VX[r][c] = OV[r][d * 16 + m] * sigm_f(OS[r][64 + m]);
    }
    __syncthreads();
  }

  // scatter into receiver segment sums
  for (int idx = tid; idx < 16 * 64; idx += 256) {
    int r = idx >> 6, c = idx & 63;
    atomicAdd(&sum_s[RIX[r] * 64 + c], SX[r][c]);
  }
  for (int idx = tid; idx < 16 * 48; idx += 256) {
    int r = idx / 48, c = idx % 48;
    atomicAdd(&sum_v[RIX[r] * 48 + c], VX[r][c]);
  }
  if (tid < 48) {
    int r = tid / 3, d = tid % 3;
    atomicAdd(&sum_av[RIX[r] * 3 + d], AV[r][d]);
  }
  if (tid < 16) atomicAdd(&cnt[RIX[tid]], 1.0f);
}

// =====================  node kernel  =====================
// 16 nodes per block, 256 threads = 8 waves. A-fragments synthesized on the
// fly from LDS; K loops split per TP region (4160 | 272, 1088 | 1040) so the
// decode is branchless.
__global__ __launch_bounds__(256) void node_kernel(
    float* __restrict__ ns, float* __restrict__ nv,
    const float* __restrict__ ysg, const float* __restrict__ yvg,
    const float* __restrict__ n_ws, const float* __restrict__ n_b,
    const float* __restrict__ n_wv, const float* __restrict__ l_ws,
    const float* __restrict__ l_b, const float* __restrict__ l_wv) {
  __shared__ float SX[16][65];   // node scalars, 64 used
  __shared__ float VX[16][49];   // node vectors flat m*3+d, 48 used
  __shared__ float YS[16][65];   // y_s (65)
  __shared__ float YV[16][51];   // y_v flat k*3+d (51)
  __shared__ float OS[16][81];
  __shared__ float OV[16][49];   // flat d*16+m

  const int tid = threadIdx.x;
  const int lane = tid & 31, wid = tid >> 5;
  const int nb = blockIdx.x * 16;

  for (int idx = tid; idx < 16 * 64; idx += 256) {
    int r = idx >> 6, c = idx & 63;
    SX[r][c] = ns[(nb + r) * 64 + c];
  }
  for (int idx = tid; idx < 16 * 48; idx += 256) {
    int r = idx / 48, c = idx % 48;
    VX[r][c] = nv[(nb + r) * 48 + c];
  }
  for (int idx = tid; idx < 16 * 65; idx += 256) {
    int r = idx / 65, c = idx % 65;
    YS[r][c] = ysg[(nb + r) * 65 + c];
  }
  for (int idx = tid; idx < 16 * 51; idx += 256) {
    int r = idx / 51, c = idx % 51;
    YV[r][c] = yvg[(nb + r) * 51 + c];
  }
  __syncthreads();

  const int mrow = lane & 15, hi = lane >> 4;
#pragma unroll 1
  for (int blk = 0; blk < 3; ++blk) {
    const float* Ws = n_ws + blk * 4432 * 80;
    const float* Wv = n_wv + blk * 2128 * 16;
    const float* Bb = n_b + blk * 80;
    if (wid < 5) {
      const int cb = wid * 16;
      const float* bp = Ws + cb + mrow;
      v8f acc = {0, 0, 0, 0, 0, 0, 0, 0};
#pragma unroll 4
      for (int k0 = 0; k0 < 4160; k0 += 4) {         // ss: t = m*65 + k
        int t = k0 + 2 * hi;
        int m = t / 65, kk = t - m * 65;
        int t1 = t + 1;
        int m1 = t1 / 65, kk1 = t1 - m1 * 65;
        float a0 = SX[mrow][m] * YS[mrow][kk];
        float a1 = SX[mrow][m1] * YS[mrow][kk1];
        acc = wmma4(a0, a1, bp[t * 80], bp[t1 * 80], acc);
      }
#pragma unroll 4
      for (int k0 = 4160; k0 < 4432; k0 += 4) {      // dd: u = m*17 + k
        int t = k0 + 2 * hi;
        int u = t - 4160;
        int m = u / 17, kk = u - m * 17;
        int u1 = u + 1;
        int m1 = u1 / 17, kk1 = u1 - m1 * 17;
        float a0 = (VX[mrow][m * 3 + 0] * YV[mrow][kk * 3 + 0] +
                    VX[mrow][m * 3 + 1] * YV[mrow][kk * 3 + 1] +
                    VX[mrow][m * 3 + 2] * YV[mrow][kk * 3 + 2]) * INV_SQRT3;
        float a1 = (VX[mrow][m1 * 3 + 0] * YV[mrow][kk1 * 3 + 0] +
                    VX[mrow][m1 * 3 + 1] * YV[mrow][kk1 * 3 + 1] +
                    VX[mrow][m1 * 3 + 2] * YV[mrow][kk1 * 3 + 2]) * INV_SQRT3;
        acc = wmma4(a0, a1, bp[t * 80], bp[(t + 1) * 80], acc);
      }
#pragma unroll
      for (int v = 0; v < 8; ++v) OS[v + 8 * hi][cb + mrow] = acc[v] + Bb[cb + mrow];
    } else {
      const int d = wid - 5;
      const float* bp = Wv + mrow;
      v8f acc = {0, 0, 0, 0, 0, 0, 0, 0};
#pragma unroll 4
      for (int k0 = 0; k0 < 1088; k0 += 4) {         // sv: t = m*17 + k
        int t = k0 + 2 * hi;
        int m = t / 17, kk = t - m * 17;
        int t1 = t + 1;
        int m1 = t1 / 17, kk1 = t1 - m1 * 17;
        float a0 = SX[mrow][m] * YV[mrow][kk * 3 + d];
        float a1 = SX[mrow][m1] * YV[mrow][kk1 * 3 + d];
        acc = wmma4(a0, a1, bp[t * 16], bp[t1 * 16], acc);
      }
#pragma unroll 4
      for (int k0 = 1088; k0 < 2128; k0 += 4) {      // vs: u = m*65 + k
        int t = k0 + 2 * hi;
        int u = t - 1088;
        int m = u / 65, kk = u - m * 65;
        int u1 = u + 1;
        int m1 = u1 / 65, kk1 = u1 - m1 * 65;
        float a0 = VX[mrow][m * 3 + d] * YS[mrow][kk];
        float a1 = VX[mrow][m1 * 3 + d] * YS[mrow][kk1];
        acc = wmma4(a0, a1, bp[t * 16], bp[(t + 1) * 16], acc);
      }
#pragma unroll
      for (int v = 0; v < 8; ++v) OV[v + 8 * hi][d * 16 + mrow] = acc[v];
    }
    __syncthreads();
    for (int idx = tid; idx < 16 * 64; idx += 256) {
      int r = idx >> 6, c = idx & 63;
      SX[r][c] = gelu_f(OS[r][c]);
    }
    for (int idx = tid; idx < 16 * 48; idx += 256) {
      int r = idx / 48, c = idx % 48;
      int m = c / 3, d = c % 3;
      VX[r][c] = OV[r][d * 16 + m] * sigm_f(OS[r][64 + m]);
    }
    __syncthreads();
  }

  // final per-step Linear (also via WMMA)
  if (wid < 4) {
    const int cb = wid * 16;
    const float* bp = l_ws + cb + mrow;
    v8f acc = {0, 0, 0, 0, 0, 0, 0, 0};
#pragma unroll 4
    for (int k0 = 0; k0 < 64; k0 += 4) {
      int k = k0 + 2 * hi;
      acc = wmma4(SX[mrow][k], SX[mrow][k + 1], bp[k * 64], bp[(k + 1) * 64], acc);
    }
#pragma unroll
    for (int v = 0; v < 8; ++v) OS[v + 8 * hi][cb + mrow] = acc[v] + l_b[cb + mrow];
  } else if (wid < 7) {
    const int d = wid - 4;
    const float* bp = l_wv + mrow;
    v8f acc = {0, 0, 0, 0, 0, 0, 0, 0};
#pragma unroll
    for (int k0 = 0; k0 < 16; k0 += 4) {
      int k = k0 + 2 * hi;
      acc = wmma4(VX[mrow][k * 3 + d], VX[mrow][(k + 1) * 3 + d], bp[k * 16],
                  bp[(k + 1) * 16], acc);
    }
#pragma unroll
    for (int v = 0; v < 8; ++v) OV[v + 8 * hi][d * 16 + mrow] = acc[v];
  }
  __syncthreads();
  for (int idx = tid; idx < 16 * 64; idx += 256) {
    int r = idx >> 6, c = idx & 63;
    ns[(nb + r) * 64 + c] = OS[r][c];
  }
  for (int idx = tid; idx < 16 * 48; idx += 256) {
    int r = idx / 48, c = idx % 48;
    nv[(nb + r) * 48 + c] = OV[r][(c % 3) * 16 + c / 3];
  }
}

// =====================  launch  =====================
extern "C" void kernel_launch(void* const* d_in, const int* in_sizes, int n_in,
                              void* d_out, int out_size, void* d_ws, size_t ws_size,
                              hipStream_t stream) {
  const float* node_s = (const float*)d_in[0];
  const float* node_v = (const float*)d_in[1];
  const int* senders = (const int*)d_in[2];
  const int* receivers = (const int*)d_in[3];
  const float* e1_ws = (const float*)d_in[4];
  const float* e1_b = (const float*)d_in[5];
  const float* e1_wv = (const float*)d_in[6];
  const float* e2_ws = (const float*)d_in[7];
  const float* e2_b = (const float*)d_in[8];
  const float* e2_wv = (const float*)d_in[9];
  const float* n_ws = (const float*)d_in[10];
  const float* n_b = (const float*)d_in[11];
  const float* n_wv = (const float*)d_in[12];
  const float* l_ws = (const float*)d_in[13];
  const float* l_b = (const float*)d_in[14];
  const float* l_wv = (const float*)d_in[15];

  float* cur_s = (float*)d_ws;                 // N*64
  float* cur_v = cur_s + NNODES * 64;          // N*48
  float* sum_s = cur_v + NNODES * 48;          // N*64, start of contiguous N*116 agg region
  float* sum_v = sum_s + NNODES * 64;          // N*48
  float* sum_av = sum_v + NNODES * 48;         // N*3
  float* cnt = sum_av + NNODES * 3;            // N, end of agg region
  float* ys = cnt + NNODES;                    // N*65
  float* yv = ys + NNODES * 65;                // N*51

  init_copy<<<(NNODES * 64 + 255) / 256, 256, 0, stream>>>(node_s, node_v, cur_s, cur_v);

  for (int step = 0; step < 3; ++step) {
    zero_buf<<<(NNODES * 116 + 255) / 256, 256, 0, stream>>>(sum_s, NNODES * 116);
    edge_kernel<<<NEDGES / 16, 256, 0, stream>>>(
        cur_s, cur_v, senders, receivers,
        e1_ws + step * 160 * 80, e1_b + step * 80, e1_wv + step * 160 * 16,
        e2_ws + step * 2 * 80 * 80, e2_b + step * 2 * 80, e2_wv + step * 2 * 80 * 16,
        sum_s, sum_v, sum_av, cnt);
    finalize_y<<<(NNODES + 127) / 128, 128, 0, stream>>>(sum_s, sum_v, sum_av, cnt, ys, yv);
    node_kernel<<<NNODES / 16, 256, 0, stream>>>(
        cur_s, cur_v, ys, yv,
        n_ws + step * 3 * 4432 * 80, n_b + step * 3 * 80, n_wv + step * 3 * 2128 * 16,
        l_ws + step * 64 * 64, l_b + step * 64, l_wv + step * 16 * 16);
  }
  writeout<<<(NNODES * 112 + 255) / 256, 256, 0, stream>>>(cur_s, cur_v, (float*)d_out);
}